// SeesawLoss_41120016892076
// MI455X (gfx1250) — compile-verified
//
#include <hip/hip_runtime.h>
#include <math.h>

// Seesaw loss (eval), N=16384 rows, C=1000 classes, p=0.6, q=1.5, eps=0.01.
// Kernel 1: one wave32 per row, row fully register-resident (32 f32/lane).
// Kernel 2: deterministic mean over 16384 row losses via V_WMMA_F32_16X16X4_F32
//           (B = ones => WMMA is a 64-wide f32 sum-reduce per issue).

#define NROWS 16384
#define NCLS  1000
#define P_EXP 0.6f
#define Q_EXP 1.5f
#define EPSW  0.01f

typedef __attribute__((ext_vector_type(2))) float v2f;
typedef __attribute__((ext_vector_type(8))) float v8f;

__device__ __forceinline__ float wave_max(float v) {
#pragma unroll
  for (int off = 16; off > 0; off >>= 1)
    v = fmaxf(v, __shfl_xor(v, off, 32));
  return v;
}
__device__ __forceinline__ float wave_sum(float v) {
#pragma unroll
  for (int off = 16; off > 0; off >>= 1)
    v += __shfl_xor(v, off, 32);
  return v;
}

// weighted logit for one element; -inf for padding lanes
__device__ __forceinline__ float wlogit(int j, float x, float cc, int tgt,
                                        float x_t, float tcnt, float m,
                                        float ln_denom) {
  if (j >= NCLS) return -INFINITY;
  float w;
  if (j == tgt) {
    w = 1.0f;
  } else {
    // mitigation: (cc_j/cc_t + eps)^p  iff cc_j < cc_t
    float mit = (cc < tcnt) ? __powf(cc / tcnt + EPSW, P_EXP) : 1.0f;
    // compensation: (p_j/(p_t+eps))^q iff p_j > p_t  <=>  x_j > x_t
    // ratio = exp(x-m)/(e_t + eps*S) -> comp = exp(q*((x-m) - ln_denom))
    float comp = (x > x_t) ? __expf(Q_EXP * ((x - m) - ln_denom)) : 1.0f;
    w = mit * comp;
  }
  return x + __logf(w + EPSW);
}

__global__ __launch_bounds__(256) void seesaw_row_loss(
    const float* __restrict__ logits, const int* __restrict__ targets,
    const float* __restrict__ cum_counts, float* __restrict__ row_loss) {
  __shared__ __align__(16) float scc[1024];  // cum_counts staged once per block
  for (int j = threadIdx.x; j < 1024; j += 256)
    scc[j] = (j < NCLS) ? cum_counts[j] : 1.0f;  // pad so tail reads are benign
  __syncthreads();

  const int lane = threadIdx.x & 31;
  const int row  = blockIdx.x * 8 + (threadIdx.x >> 5);
  const long base = (long)row * NCLS;

  const int   tgt  = targets[row];
  const float x_t  = logits[base + tgt];          // uniform broadcast load
  const float tcnt = fmaxf(scc[tgt], 1.0f);

  // ---- pass 1: load row (coalesced b128), row max, S = sum exp(x-m) ----
  float4 v[8];
  float m = -INFINITY;
#pragma unroll
  for (int it = 0; it < 8; ++it) {
    const int j0 = it * 128 + lane * 4;           // 1000 = 7*128 + 26*4 (clean tail)
    if (j0 < NCLS) {
      v[it] = *(const float4*)(logits + base + j0);
    } else {
      v[it] = make_float4(-INFINITY, -INFINITY, -INFINITY, -INFINITY);
    }
    m = fmaxf(m, fmaxf(fmaxf(v[it].x, v[it].y), fmaxf(v[it].z, v[it].w)));
  }
  m = wave_max(m);

  float s = 0.0f;
#pragma unroll
  for (int it = 0; it < 8; ++it) {
    s += __expf(v[it].x - m);   // exp(-inf)=0 for padding
    s += __expf(v[it].y - m);
    s += __expf(v[it].z - m);
    s += __expf(v[it].w - m);
  }
  s = wave_sum(s);

  const float denom    = __expf(x_t - m) + EPSW * s;  // e_t + eps*S
  const float ln_denom = __logf(denom);

  // ---- pass 2 (register resident): weighted logits + logsumexp ----
  float m2 = -INFINITY;
#pragma unroll
  for (int it = 0; it < 8; ++it) {
    const int j0 = it * 128 + lane * 4;           // scc read always in [0,1024)
    const float4 cc = *(const float4*)(scc + j0);
    float4 w;
    w.x = wlogit(j0 + 0, v[it].x, cc.x, tgt, x_t, tcnt, m, ln_denom);
    w.y = wlogit(j0 + 1, v[it].y, cc.y, tgt, x_t, tcnt, m, ln_denom);
    w.z = wlogit(j0 + 2, v[it].z, cc.z, tgt, x_t, tcnt, m, ln_denom);
    w.w = wlogit(j0 + 3, v[it].w, cc.w, tgt, x_t, tcnt, m, ln_denom);
    v[it] = w;
    m2 = fmaxf(m2, fmaxf(fmaxf(w.x, w.y), fmaxf(w.z, w.w)));
  }
  m2 = wave_max(m2);

  float s2 = 0.0f;
#pragma unroll
  for (int it = 0; it < 8; ++it) {
    s2 += __expf(v[it].x - m2);
    s2 += __expf(v[it].y - m2);
    s2 += __expf(v[it].z - m2);
    s2 += __expf(v[it].w - m2);
  }
  s2 = wave_sum(s2);

  // loss = logsumexp(wl) - wl[target];  wl[t] = x_t + log(1+eps)
  const float loss = (m2 + __logf(s2)) - (x_t + __logf(1.0f + EPSW));
  if (lane == 0) row_loss[row] = loss;
}

// ---- deterministic mean of 16384 floats via WMMA (ones-matrix reduce) ----
__global__ __launch_bounds__(256) void wmma_mean(const float* __restrict__ in,
                                                 float* __restrict__ out) {
  __shared__ float partial[8];
  const int lane = threadIdx.x & 31;
  const int wave = threadIdx.x >> 5;

  v2f ones;  ones.x = 1.0f;  ones.y = 1.0f;   // B = 4x16 all-ones
  v8f c = {};                                  // f32 16x16 accumulator

  // each WMMA consumes a 64-element chunk (A = 16x4 f32); 8 waves x 32 iters
#pragma unroll 4
  for (int it = 0; it < (NROWS / 64) / 8; ++it) {
    const int chunk = it * 8 + wave;
    const v2f a = *(const v2f*)(in + chunk * 64 + lane * 2);  // coalesced b64
    // D[m][n] = sum_k A[m][k] * 1  -> per-row sums, accumulated into C
    c = __builtin_amdgcn_wmma_f32_16x16x4_f32(
        /*neg_a=*/false, a, /*neg_b=*/false, ones,
        /*c_mod=*/(short)0, c, /*reuse_a=*/false, /*reuse_b=*/false);
  }
  // lanes 0-15 hold rows 0-7 in c[0..7]; lanes 16-31 hold rows 8-15
  float s = c[0] + c[1] + c[2] + c[3] + c[4] + c[5] + c[6] + c[7];
  s += __shfl_xor(s, 16, 32);                 // combine the two half-wave groups
  if (lane == 0) partial[wave] = s;
  __syncthreads();
  if (threadIdx.x == 0) {
    float t = 0.0f;
#pragma unroll
    for (int w = 0; w < 8; ++w) t += partial[w];
    out[0] = t * (1.0f / (float)NROWS);
  }
}

extern "C" void kernel_launch(void* const* d_in, const int* in_sizes, int n_in,
                              void* d_out, int out_size, void* d_ws,
                              size_t ws_size, hipStream_t stream) {
  const float* logits     = (const float*)d_in[0];
  const int*   targets    = (const int*)d_in[1];
  const float* cum_counts = (const float*)d_in[2];
  float*       row_loss   = (float*)d_ws;     // 16384 floats = 64 KB scratch

  seesaw_row_loss<<<NROWS / 8, 256, 0, stream>>>(logits, targets, cum_counts,
                                                 row_loss);
  wmma_mean<<<1, 256, 0, stream>>>(row_loss, (float*)d_out);
}